// RK4PointMassSimulator_38671885534077
// MI455X (gfx1250) — compile-verified
//
#include <hip/hip_runtime.h>
#include <hip/hip_bf16.h>

#define N_ENV   65536
#define T_STEPS 256
#define T_C     16
#define NCH     (T_STEPS / T_C)
#define BLOCK   256
#define NWAVE   (BLOCK / 32)
#define DT_F    (1.0f / 30.0f)

#if defined(__has_builtin)
#if __has_builtin(__builtin_amdgcn_global_load_async_to_lds_b64)
#define HAVE_ASYNC 1
#endif
#endif

typedef int   v2i __attribute__((ext_vector_type(2)));
typedef float v4f __attribute__((ext_vector_type(4)));
typedef __attribute__((address_space(1))) v2i Gv2i;  // global, 64-bit payload
typedef __attribute__((address_space(3))) v2i Lv2i;  // LDS, 64-bit payload

// accel = thrust * vel_dir + torque * perp(vel_dir); perp = (-dy, dx)
__device__ __forceinline__ float2 dynamics_acc(float vx, float vy, float th, float tq) {
    float n2  = fmaf(vx, vx, vy * vy);
    // max(sqrt(n2), 1e-6) == sqrt(max(n2, 1e-12)); one v_rsq_f32 instead of sqrt+div
    float inv = __builtin_amdgcn_rsqf(fmaxf(n2, 1e-12f));
    float dx = vx * inv;
    float dy = vy * inv;
    return make_float2(fmaf(th, dx, -(tq * dy)), fmaf(th, dy, tq * dx));
}

__global__ __launch_bounds__(BLOCK)
void rk4_rollout_kernel(const float* __restrict__ init,
                        const float2* __restrict__ act,
                        float* __restrict__ out) {
#if HAVE_ASYNC
    // [buf][wave][step][envLocal] -> compute read stride across lanes = 8B: conflict-free
    __shared__ float2 ldsA[2][NWAVE][T_C][32];
#endif
    const int tid  = threadIdx.x;
    const int wave = tid >> 5;
    const int lane = tid & 31;
    const int env  = blockIdx.x * BLOCK + tid;

#if HAVE_ASYNC
    const int envW = blockIdx.x * BLOCK + wave * 32;  // first env owned by this wave
    // Wave stages its own 32 envs x T_C steps. Lane l covers env (l>>4)+2i, step l&15.
    // Global side per instruction: two contiguous 128B rows -> fully coalesced.
    auto issue_chunk = [&](int c, int b) {
        const float2* g0 = act + (size_t)(envW + (lane >> 4)) * T_STEPS
                               + (size_t)c * T_C + (lane & 15);
        float2* l0 = &ldsA[b][wave][lane & 15][lane >> 4];
#pragma unroll
        for (int i = 0; i < 16; ++i) {
            __builtin_amdgcn_global_load_async_to_lds_b64(
                (Gv2i*)(g0 + (size_t)i * 2 * T_STEPS),  // env += 2 per i
                (Lv2i*)(l0 + i * 2),                    // envLocal += 2 per i
                0, 0);
        }
    };
    issue_chunk(0, 0);
#endif

    const float4 s = reinterpret_cast<const float4*>(init)[env];
    float px = s.x, py = s.y, vx = s.z, vy = s.w;
    const float dt  = DT_F;
    const float hdt = 0.5f * DT_F;
    const float sdt = DT_F / 6.0f;
    v4f* __restrict__ orow = reinterpret_cast<v4f*>(out) + (size_t)env * T_STEPS;

    for (int c = 0; c < NCH; ++c) {
#if HAVE_ASYNC
        if (c + 1 < NCH) {
            issue_chunk(c + 1, (c + 1) & 1);
            // in-order async completion: <=16 outstanding => chunk c fully in LDS
            asm volatile("s_wait_asynccnt 0x10" ::: "memory");
        } else {
            asm volatile("s_wait_asynccnt 0x0" ::: "memory");
        }
#endif
        for (int t = 0; t < T_C; ++t) {
#if HAVE_ASYNC
            const float2 a = ldsA[c & 1][wave][t][lane];
#else
            const float2 a = act[(size_t)env * T_STEPS + c * T_C + t];
#endif
            const float th = a.x, tq = a.y;

            // RK4: position derivative is the stage velocity; accel depends on vel only
            float2 a1 = dynamics_acc(vx, vy, th, tq);
            float v2x = fmaf(hdt, a1.x, vx), v2y = fmaf(hdt, a1.y, vy);
            float2 a2 = dynamics_acc(v2x, v2y, th, tq);
            float v3x = fmaf(hdt, a2.x, vx), v3y = fmaf(hdt, a2.y, vy);
            float2 a3 = dynamics_acc(v3x, v3y, th, tq);
            float v4x = fmaf(dt, a3.x, vx), v4y = fmaf(dt, a3.y, vy);
            float2 a4 = dynamics_acc(v4x, v4y, th, tq);

            px = fmaf(sdt, vx + 2.0f * (v2x + v3x) + v4x, px);
            py = fmaf(sdt, vy + 2.0f * (v2y + v3y) + v4y, py);
            vx = fmaf(sdt, a1.x + 2.0f * (a2.x + a3.x) + a4.x, vx);
            vy = fmaf(sdt, a1.y + 2.0f * (a2.y + a3.y) + a4.y, vy);

            // 256 MB streaming output, never re-read: non-temporal b128 store
            v4f o = {px, py, vx, vy};
            __builtin_nontemporal_store(o, orow + c * T_C + t);
        }
    }
}

extern "C" void kernel_launch(void* const* d_in, const int* in_sizes, int n_in,
                              void* d_out, int out_size, void* d_ws, size_t ws_size,
                              hipStream_t stream) {
    const float*  init = (const float*)d_in[0];   // (65536, 4) f32
    const float2* act  = (const float2*)d_in[1];  // (65536, 256, 2) f32
    float*        out  = (float*)d_out;           // (65536, 256, 4) f32
    rk4_rollout_kernel<<<dim3(N_ENV / BLOCK), dim3(BLOCK), 0, stream>>>(init, act, out);
}